// CombineValuesLayer_50491635532429
// MI455X (gfx1250) — compile-verified
//
#include <hip/hip_runtime.h>
#include <stdint.h>

#define DLEN 8192
#define NT 256
#define F4T (DLEN / 4 / NT)   // float4s per thread = 8

typedef float vfloat4 __attribute__((ext_vector_type(4)));

// Order-preserving float -> uint key (ascending uint == ascending float)
__device__ __forceinline__ uint32_t f2key(float f) {
  uint32_t u = __float_as_uint(f);
  return u ^ ((uint32_t)((int32_t)u >> 31) | 0x80000000u);
}
__device__ __forceinline__ float key2f(uint32_t k) {
  uint32_t u = (k & 0x80000000u) ? (k ^ 0x80000000u) : ~k;
  return __uint_as_float(u);
}

// Wave-0-only 256-bin descending selection: find bin b with
// sufsum[b] > r >= sufsum[b+1]  (sufsum = suffix sum of hist),
// publish bin and the residual rank. No workgroup barriers inside.
__device__ __forceinline__ void select_bin(const uint32_t* hist, uint32_t r,
                                           uint32_t* s_bin, uint32_t* s_rank,
                                           int t) {
  if (t < 32) {
    uint32_t h[8];
    uint32_t S = 0;
#pragma unroll
    for (int b = 0; b < 8; ++b) { h[b] = hist[t * 8 + b]; S += h[b]; }
    // inclusive suffix-sum across the 32 lanes (wave32, 5 shuffle steps)
    uint32_t incl = S;
#pragma unroll
    for (int d = 1; d < 32; d <<= 1) {
      uint32_t o = __shfl_down(incl, d, 32);
      if (t + d < 32) incl += o;
    }
    uint32_t above = incl - S;  // sum of all bins in lanes > t
    // walk this lane's 8 bins from high to low
#pragma unroll
    for (int b = 7; b >= 0; --b) {
      uint32_t cur = above + h[b];              // sufsum[t*8 + b]
      if (cur > r && above <= r) {              // exactly one (lane,b) hits
        *s_bin = (uint32_t)(t * 8 + b);
        *s_rank = r - above;
      }
      above = cur;
    }
  }
}

__global__ __launch_bounds__(NT) void combine_topk_kernel(
    const float* __restrict__ A, const float* __restrict__ B,
    const int* __restrict__ thrp, float* __restrict__ C) {
  __shared__ __align__(16) float sA[DLEN];
  __shared__ __align__(16) float sB[DLEN];
  __shared__ __align__(16) uint32_t cand[DLEN];
  __shared__ uint32_t hist[NT];
  __shared__ uint32_t s_bin, s_rank, s_cnt;

  const int t = threadIdx.x;
  const size_t row = blockIdx.x;
  const float* ga = A + row * (size_t)DLEN;
  const float* gb = B + row * (size_t)DLEN;
  float* gc = C + row * (size_t)DLEN;

  // ---------------- async DMA: both rows global -> LDS (ASYNCcnt path) ----
  {
    uint32_t ldsA = (uint32_t)(uintptr_t)&sA[0];
    uint32_t ldsB = (uint32_t)(uintptr_t)&sB[0];
    uint32_t lane = (uint32_t)t * 16u;
#pragma unroll
    for (int j = 0; j < 8; ++j) {
      uint32_t o = lane + (uint32_t)j * 4096u;
      asm volatile("global_load_async_to_lds_b128 %0, %1, %2"
                   :: "v"(ldsA + o), "v"(o), "s"(ga) : "memory");
      asm volatile("global_load_async_to_lds_b128 %0, %1, %2"
                   :: "v"(ldsB + o), "v"(o), "s"(gb) : "memory");
    }
    asm volatile("s_wait_asynccnt 0" ::: "memory");
  }
  __syncthreads();

  const uint32_t rank0 = (uint32_t)(*thrp);  // k-th from top, 0-indexed (=32)

  float th[2];
#pragma unroll
  for (int which = 0; which < 2; ++which) {
    const vfloat4* v4 = (const vfloat4*)(which ? sB : sA);

    // ---- pass 0: 256-bin histogram of top key byte over the full row ----
    hist[t] = 0;
    if (t == 0) s_cnt = 0;
    __syncthreads();
#pragma unroll
    for (int j = 0; j < F4T; ++j) {
      vfloat4 x = v4[t + j * NT];
#pragma unroll
      for (int c = 0; c < 4; ++c)
        atomicAdd(&hist[f2key(x[c]) >> 24], 1u);
    }
    __syncthreads();
    select_bin(hist, rank0, &s_bin, &s_rank, t);
    __syncthreads();
    uint32_t pfx = s_bin;

    // ---- compact keys whose top byte matches (expected ~hundreds) ----
#pragma unroll
    for (int j = 0; j < F4T; ++j) {
      vfloat4 x = v4[t + j * NT];
#pragma unroll
      for (int c = 0; c < 4; ++c) {
        uint32_t u = f2key(x[c]);
        if ((u >> 24) == pfx) cand[atomicAdd(&s_cnt, 1u)] = u;
      }
    }
    __syncthreads();
    const int m = (int)s_cnt;

    // ---- passes 1..3 over candidate list only ----
#pragma unroll 1
    for (int p = 1; p < 4; ++p) {
      const int shift = 24 - 8 * p;
      uint32_t r = s_rank;   // write is barrier-separated from this read
      hist[t] = 0;
      __syncthreads();
      for (int i = t; i < m; i += NT) {
        uint32_t u = cand[i];
        if ((u >> (shift + 8)) == pfx)
          atomicAdd(&hist[(u >> shift) & 0xFFu], 1u);
      }
      __syncthreads();
      select_bin(hist, r, &s_bin, &s_rank, t);
      __syncthreads();
      pfx = (pfx << 8) | s_bin;
    }
    th[which] = key2f(pfx);   // exact bit pattern of the k-th largest
    __syncthreads();          // protect s_* reuse by next iteration
  }

  const float mv = fmaxf(th[0], th[1]);

  // ---- elementwise combine from LDS, streaming (NT) stores to global ----
  const vfloat4* a4 = (const vfloat4*)sA;
  const vfloat4* b4 = (const vfloat4*)sB;
  vfloat4* c4 = (vfloat4*)gc;
#pragma unroll
  for (int j = 0; j < F4T; ++j) {
    int i = t + j * NT;
    vfloat4 a = a4[i];
    vfloat4 b = b4[i];
    vfloat4 r;
#pragma unroll
    for (int c = 0; c < 4; ++c) {
      bool keep = (a[c] <= mv) || (b[c] <= mv);
      r[c] = keep ? a[c] * b[c] : a[c] + b[c];
    }
    __builtin_nontemporal_store(r, &c4[i]);
  }
}

extern "C" void kernel_launch(void* const* d_in, const int* in_sizes, int n_in,
                              void* d_out, int out_size, void* d_ws, size_t ws_size,
                              hipStream_t stream) {
  const float* A = (const float*)d_in[0];
  const float* B = (const float*)d_in[1];
  const int* thr = (const int*)d_in[2];
  float* C = (float*)d_out;
  int rows = in_sizes[0] / DLEN;   // 4*2048 = 8192 rows
  combine_topk_kernel<<<dim3(rows), dim3(NT), 0, stream>>>(A, B, thr, C);
}